// ProtNgramGCN_77309412201
// MI455X (gfx1250) — compile-verified
//
#include <hip/hip_runtime.h>
#include <hip/hip_bf16.h>
#include <math.h>

// ---------------------------------------------------------------------------
// ProtNgramGCN on MI455X (gfx1250, wave32)
//   h0 = x + pe                                      (elementwise)
//   per layer:  Wc = Wmain + Wshared  (halves propagation passes, linearity)
//               Y  = h @ Wc           (V_WMMA_F32_16X16X4_F32 GEMM)
//               acc = res + cin*bIn + cout*bOut
//               acc[dst] += cin/cout[dst] * ew * Y[src]   (L2-resident atomics)
//               h' = tanh(acc)
//   logits = h2 @ decW + dec_b ; logp = log_softmax ; emb = h2/(||h2||+eps)
// ---------------------------------------------------------------------------

typedef __attribute__((ext_vector_type(2))) float v2f;
typedef __attribute__((ext_vector_type(8))) float v8f;

// ---------------- elementwise ----------------
__global__ void add_pe_kernel(const float* __restrict__ x, const float* __restrict__ pe,
                              float* __restrict__ h, int total) {
  int i = blockIdx.x * blockDim.x + threadIdx.x;
  if (i < total) h[i] = x[i] + pe[i & 127];   // pe flat[g*32+o] == c for c<128
}

__global__ void add2_kernel(const float* __restrict__ a, const float* __restrict__ b,
                            float* __restrict__ o, int n) {
  int i = blockIdx.x * blockDim.x + threadIdx.x;
  if (i < n) o[i] = a[i] + b[i];
}

__global__ void init_acc_kernel(const float* __restrict__ res,
                                const float* __restrict__ cin, const float* __restrict__ cout,
                                const float* __restrict__ bIn, const float* __restrict__ bOut,
                                float* __restrict__ acc, int dim, int total) {
  int i = blockIdx.x * blockDim.x + threadIdx.x;
  if (i >= total) return;
  int n = i / dim, c = i - n * dim;
  acc[i] = res[i] + cin[n] * bIn[c] + cout[n] * bOut[c];
}

__global__ void tanh_inplace_kernel(float* __restrict__ a, int n) {
  int i = blockIdx.x * blockDim.x + threadIdx.x;
  if (i < n) a[i] = tanhf(a[i]);
}

// ---------------- WMMA f32 GEMM:  Y[M,Ndo] = A[M,K] @ W[K,Ndo] (+ bias) ----
// One wave computes one 16x16 tile, accumulating over K in steps of 4 via
// V_WMMA_F32_16X16X4_F32.  Layouts per CDNA5 ISA 7.12.2:
//   A 16x4 : lane<16 -> row=lane,    K={0,1} in v0,v1 ; lane>=16 -> K={2,3}
//   B 4x16 : v0: lanes0-15 row K=0 / lanes16-31 K=2 ; v1: K=1 / K=3
//   C 16x16: vgpr v, lanes0-15 -> (M=v, N=lane); lanes16-31 -> (M=8+v)
__global__ void gemm_wmma_f32(const float* __restrict__ A, const float* __restrict__ W,
                              const float* __restrict__ bias, float* __restrict__ Y,
                              int M, int K, int Ndo) {
  int lane   = threadIdx.x & 31;
  int waveId = (blockIdx.x * blockDim.x + threadIdx.x) >> 5;
  int colTiles = Ndo >> 4;
  int rowTiles = (M + 15) >> 4;
  if (waveId >= rowTiles * colTiles) return;          // wave-uniform exit
  int rowTile = waveId / colTiles;
  int colTile = waveId - rowTile * colTiles;
  int rowBase = rowTile << 4;
  int colBase = colTile << 4;
  int col  = colBase + (lane & 15);
  int rowA = rowBase + (lane & 15);
  if (rowA >= M) rowA = M - 1;                        // clamp: EXEC stays all-ones
  int koff = (lane >> 4) << 1;                        // 0 or 2
  const float* arow = A + (size_t)rowA * K;

  v8f c = {0.f, 0.f, 0.f, 0.f, 0.f, 0.f, 0.f, 0.f};
  for (int k = 0; k < K; k += 4) {
    v2f a, b;
    a.x = arow[k + koff];
    a.y = arow[k + koff + 1];
    b.x = W[(size_t)(k + koff)     * Ndo + col];
    b.y = W[(size_t)(k + koff + 1) * Ndo + col];
    c = __builtin_amdgcn_wmma_f32_16x16x4_f32(false, a, false, b, (short)0, c, false, false);
  }

  float bv = bias ? bias[col] : 0.0f;
  int mBase = rowBase + ((lane >> 4) << 3);
#pragma unroll
  for (int v = 0; v < 8; ++v) {
    int r = mBase + v;
    if (r < M) Y[(size_t)r * Ndo + col] = c[v] + bv;
  }
}

// ---------------- fused edge propagation -----------------------------------
// acc[dst,:] += cscale[dst] * ew[e] * Y[src,:]   (one wave per edge; rows are
// coalesced 128B reads per wave; atomics hit L2-resident accumulator)
__global__ void scatter_edges_kernel(const long long* __restrict__ ei, // [2,E]
                                     const float* __restrict__ ew,
                                     const float* __restrict__ Y,
                                     const float* __restrict__ cscale,
                                     float* __restrict__ acc, int dim, int E) {
  int tid    = blockIdx.x * blockDim.x + threadIdx.x;
  int wid    = tid >> 5;
  int lane   = tid & 31;
  int nWaves = (gridDim.x * blockDim.x) >> 5;
  for (int e = wid; e < E; e += nWaves) {
    int src = (int)ei[e];
    int dst = (int)ei[(size_t)E + e];
    float s = ew[e] * cscale[dst];
    const float* ys = Y   + (size_t)src * dim;
    float*       ad = acc + (size_t)dst * dim;
    for (int d = lane; d < dim; d += 32)
      atomicAdd(&ad[d], s * ys[d]);
  }
}

// ---------------- wave reductions ----------------
__device__ inline float wave_max(float v) {
  for (int o = 16; o > 0; o >>= 1) v = fmaxf(v, __shfl_xor(v, o, 32));
  return v;
}
__device__ inline float wave_sum(float v) {
  for (int o = 16; o > 0; o >>= 1) v += __shfl_xor(v, o, 32);
  return v;
}

// one wave per row of 128 logits
__global__ void logsoftmax128_kernel(const float* __restrict__ logits,
                                     float* __restrict__ out, int n) {
  int wave = (blockIdx.x * blockDim.x + threadIdx.x) >> 5;
  int lane = threadIdx.x & 31;
  if (wave >= n) return;
  const float* row = logits + (size_t)wave * 128;
  float v0 = row[lane], v1 = row[lane + 32], v2 = row[lane + 64], v3 = row[lane + 96];
  float m = wave_max(fmaxf(fmaxf(v0, v1), fmaxf(v2, v3)));
  float s = wave_sum(expf(v0 - m) + expf(v1 - m) + expf(v2 - m) + expf(v3 - m));
  float ls = m + logf(s);
  float* orow = out + (size_t)wave * 128;
  orow[lane]      = v0 - ls;
  orow[lane + 32] = v1 - ls;
  orow[lane + 64] = v2 - ls;
  orow[lane + 96] = v3 - ls;
}

// one wave per row of 64: emb = h / (||h|| + 1e-12)
__global__ void l2norm64_kernel(const float* __restrict__ h, float* __restrict__ out, int n) {
  int wave = (blockIdx.x * blockDim.x + threadIdx.x) >> 5;
  int lane = threadIdx.x & 31;
  if (wave >= n) return;
  const float* row = h + (size_t)wave * 64;
  float a = row[lane], b = row[lane + 32];
  float ss = wave_sum(a * a + b * b);
  float inv = 1.0f / (sqrtf(ss) + 1e-12f);
  float* orow = out + (size_t)wave * 64;
  orow[lane]      = a * inv;
  orow[lane + 32] = b * inv;
}

// ---------------------------------------------------------------------------
static inline int cdiv(int a, int b) { return (a + b - 1) / b; }

extern "C" void kernel_launch(void* const* d_in, const int* in_sizes, int n_in,
                              void* d_out, int out_size, void* d_ws, size_t ws_size,
                              hipStream_t stream) {
  (void)n_in; (void)out_size; (void)ws_size;

  const float*     x      = (const float*)d_in[0];
  const long long* ei_in  = (const long long*)d_in[1];
  const float*     ew_in  = (const float*)d_in[2];
  const long long* ei_out = (const long long*)d_in[3];
  const float*     ew_out = (const float*)d_in[4];
  const float*     pe     = (const float*)d_in[5];

  // layer param blocks (9 tensors each, starting at index 6)
  const float* Wmi[2] = {(const float*)d_in[6],  (const float*)d_in[15]};
  const float* Wmo[2] = {(const float*)d_in[7],  (const float*)d_in[16]};
  const float* Ws [2] = {(const float*)d_in[8],  (const float*)d_in[17]};
  const float* bmi[2] = {(const float*)d_in[9],  (const float*)d_in[18]};
  const float* bmo[2] = {(const float*)d_in[10], (const float*)d_in[19]};
  const float* bsi[2] = {(const float*)d_in[11], (const float*)d_in[20]};
  const float* bso[2] = {(const float*)d_in[12], (const float*)d_in[21]};
  const float* cin[2] = {(const float*)d_in[13], (const float*)d_in[22]};
  const float* cou[2] = {(const float*)d_in[14], (const float*)d_in[23]};
  const float* resW = (const float*)d_in[24];
  const float* resB = (const float*)d_in[25];
  const float* decW = (const float*)d_in[26];
  const float* decB = (const float*)d_in[27];

  const int N = in_sizes[0] / 128;
  const int E = in_sizes[2];
  const int DI[2] = {128, 128};
  const int DO[2] = {128, 64};

  // ---- workspace layout (floats) ----
  float* ws = (float*)d_ws;
  size_t o = 0;
  float* h0    = ws + o; o += (size_t)N * 128;
  float* Y     = ws + o; o += (size_t)N * 128;   // GEMM output, propagated
  float* acc0  = ws + o; o += (size_t)N * 128;   // layer-0 accumulator -> h1
  float* res1  = ws + o; o += (size_t)N * 64;
  float* acc1  = ws + o; o += (size_t)N * 64;    // layer-1 accumulator -> h2
  float* logit = ws + o; o += (size_t)N * 128;
  float* WcIn  = ws + o; o += 128 * 128;
  float* WcOut = ws + o; o += 128 * 128;
  float* bIn   = ws + o; o += 128;
  float* bOut  = ws + o; o += 128;

  const int T = 256;

  // ---- positional encoding ----
  add_pe_kernel<<<cdiv(N * 128, T), T, 0, stream>>>(x, pe, h0, N * 128);

  float* hcur = h0;
  float* accs[2] = {acc0, acc1};
  for (int l = 0; l < 2; ++l) {
    const int di = DI[l], dn = DO[l];
    float* acc = accs[l];

    // combined weights/biases (linearity of propagation: halves scatter passes)
    add2_kernel<<<cdiv(di * dn, T), T, 0, stream>>>(Wmi[l], Ws[l], WcIn,  di * dn);
    add2_kernel<<<cdiv(di * dn, T), T, 0, stream>>>(Wmo[l], Ws[l], WcOut, di * dn);
    add2_kernel<<<1, 128, 0, stream>>>(bmi[l], bsi[l], bIn,  dn);
    add2_kernel<<<1, 128, 0, stream>>>(bmo[l], bso[l], bOut, dn);

    // residual for this layer
    const float* res = hcur;                 // layer 0: identity residual
    if (l == 1) {
      int waves = cdiv(N, 16) * (64 >> 4);
      gemm_wmma_f32<<<cdiv(waves * 32, T), T, 0, stream>>>(hcur, resW, resB, res1, N, 128, 64);
      res = res1;
    }

    // acc = res + cin*bIn + cout*bOut
    init_acc_kernel<<<cdiv(N * dn, T), T, 0, stream>>>(res, cin[l], cou[l], bIn, bOut,
                                                       acc, dn, N * dn);

    int waves = cdiv(N, 16) * (dn >> 4);
    int gblk  = cdiv(waves * 32, T);
    int sblk  = 2048;                        // grid-stride scatter, 16k waves

    // in-channel: Y = h @ (Wmi+Ws); acc[dst] += cin[dst]*ew*Y[src]
    gemm_wmma_f32<<<gblk, T, 0, stream>>>(hcur, WcIn, nullptr, Y, N, di, dn);
    scatter_edges_kernel<<<sblk, T, 0, stream>>>(ei_in, ew_in, Y, cin[l], acc, dn, E);

    // out-channel: Y = h @ (Wmo+Ws); acc[dst] += cout[dst]*ew*Y[src]
    gemm_wmma_f32<<<gblk, T, 0, stream>>>(hcur, WcOut, nullptr, Y, N, di, dn);
    scatter_edges_kernel<<<sblk, T, 0, stream>>>(ei_out, ew_out, Y, cou[l], acc, dn, E);

    // h' = tanh(acc)  (in place; acc becomes the next layer input)
    tanh_inplace_kernel<<<cdiv(N * dn, T), T, 0, stream>>>(acc, N * dn);
    hcur = acc;
  }

  // ---- decoder + heads ----
  float* out  = (float*)d_out;
  float* logp = out;                       // [N,128]
  float* emb  = out + (size_t)N * 128;     // [N,64]

  int dwaves = cdiv(N, 16) * (128 >> 4);
  gemm_wmma_f32<<<cdiv(dwaves * 32, T), T, 0, stream>>>(hcur, decW, decB, logit, N, 64, 128);

  logsoftmax128_kernel<<<cdiv(N * 32, T), T, 0, stream>>>(logit, logp, N);
  l2norm64_kernel<<<cdiv(N * 32, T), T, 0, stream>>>(hcur, emb, N);
}